// stepSSM_62285615727372
// MI455X (gfx1250) — compile-verified
//
#include <hip/hip_runtime.h>
#include <cstdint>
#include <cstddef>

typedef float v4f __attribute__((ext_vector_type(4)));
typedef float v2f __attribute__((ext_vector_type(2)));

#define NEG_SLOPE 0.125f

// LDS weight layout (floats):
//  [  0, 64)  A     : i*16 + h*8 + f*2 + c
//  [ 64,128)  Bm    : i*16 + h*8 + f*2 + c
//  [128,192)  C     : i*16 + h*8 + f*2 + c   (c-dim of C is 1, dropped)
//  [192,200)  D     : i*2 + h
//  [200,208)  fc1_w : c*4 + k
//  [208,210)  fc1_b : c
//  [210,230)  fc_w  : i*4 + c*2 + k
//  [230,240)  fc_b  : i*2 + c
__global__ __launch_bounds__(256) void ssm_chain_kernel(
    const float* __restrict__ x,      // [N,4]
    const float* __restrict__ st,     // [4,N,2,4,2]
    const float* __restrict__ fc1w,   // [2,4]
    const float* __restrict__ fc1b,   // [2]
    const float* __restrict__ fcw,    // [5,2,2]
    const float* __restrict__ fcb,    // [5,2]
    const float* __restrict__ Aw,     // [4,2,4,2]
    const float* __restrict__ Bw,     // [4,2,4,2]
    const float* __restrict__ Cw,     // [4,1,2,4,2]
    const float* __restrict__ Dw,     // [4,1,2]
    float* __restrict__ out_h,        // [N,2]
    float* __restrict__ out_s,        // [4,N,2,4,2]
    int N)
{
    __shared__ float W[240];

    const int tid = threadIdx.x;
    const int n   = blockIdx.x * blockDim.x + tid;
    const int nc  = (n < N) ? n : (N - 1);      // clamped: all threads reach barrier

    // ---- issue ALL streaming loads first (LOADcnt rides across the barrier;
    //      s_wait_asynccnt / s_barrier below do not drain LOADcnt) ----------
    v4f xv = __builtin_nontemporal_load((const v4f*)(x + (size_t)nc * 4));

    const size_t nb = (size_t)N * 16;           // floats per layer slab
    const float* sp = st + (size_t)nc * 16;
    float sf[64];
    #pragma unroll
    for (int i = 0; i < 4; ++i) {
        #pragma unroll
        for (int q = 0; q < 4; ++q) {
            v4f t4 = __builtin_nontemporal_load(
                         (const v4f*)(sp + (size_t)i * nb + q * 4));
            sf[i*16 + q*4 + 0] = t4.x;
            sf[i*16 + q*4 + 1] = t4.y;
            sf[i*16 + q*4 + 2] = t4.z;
            sf[i*16 + q*4 + 3] = t4.w;
        }
    }

    // ---- CDNA5 async global->LDS staging of all weights (ASYNCcnt path) ----
    {
        const float* src = nullptr;
        if      (tid < 64)  src = Aw   + tid;
        else if (tid < 128) src = Bw   + (tid - 64);
        else if (tid < 192) src = Cw   + (tid - 128);
        else if (tid < 200) src = Dw   + (tid - 192);
        else if (tid < 208) src = fc1w + (tid - 200);
        else if (tid < 210) src = fc1b + (tid - 208);
        else if (tid < 230) src = fcw  + (tid - 210);
        else if (tid < 240) src = fcb  + (tid - 230);
        if (src) {
            // low 32 bits of the flat shared pointer == wave-relative LDS byte offset
            uint32_t lds_off = (uint32_t)(uintptr_t)(&W[tid]);
            uint64_t gaddr   = (uint64_t)(uintptr_t)src;
            asm volatile("global_load_async_to_lds_b32 %0, %1, off"
                         :: "v"(lds_off), "v"(gaddr) : "memory");
        }
        asm volatile("s_wait_asynccnt 0" ::: "memory");
        __syncthreads();
    }

    if (n >= N) return;

    // ---- fc1: [4] -> [2] ----
    float h0 = W[208] + xv.x*W[200] + xv.y*W[201] + xv.z*W[202] + xv.w*W[203];
    float h1 = W[209] + xv.x*W[204] + xv.y*W[205] + xv.z*W[206] + xv.w*W[207];

    float* op = out_s + (size_t)n * 16;

    // ---- 4 x (SSM step -> leaky_relu -> fc) ----
    #pragma unroll
    for (int i = 0; i < 4; ++i) {
        const float u0 = h0, u1 = h1;
        float y[2];
        float nsf[16];
        #pragma unroll
        for (int hh = 0; hh < 2; ++hh) {
            const float uu = (hh == 0) ? u0 : u1;
            float acc = 0.0f;
            #pragma unroll
            for (int f = 0; f < 4; ++f) {
                const int wi = i*16 + hh*8 + f*2;
                const float sr = sf[i*16 + hh*8 + f*2 + 0];
                const float si = sf[i*16 + hh*8 + f*2 + 1];
                const float ar = W[      wi], ai = W[      wi + 1];
                const float br = W[ 64 + wi], bi = W[ 64 + wi + 1];
                const float cr = W[128 + wi], ci = W[128 + wi + 1];
                const float nr = fmaf(ar, sr, fmaf(-ai, si, br * uu));
                const float ni = fmaf(ar, si, fmaf( ai, sr, bi * uu));
                nsf[hh*8 + f*2 + 0] = nr;
                nsf[hh*8 + f*2 + 1] = ni;
                acc = fmaf(cr, nr, fmaf(-ci, ni, acc));
            }
            y[hh] = fmaf(2.0f, acc, uu * W[192 + i*2 + hh]);
        }
        // stream out new state (non-temporal b128)
        #pragma unroll
        for (int q = 0; q < 4; ++q) {
            v4f v = { nsf[q*4+0], nsf[q*4+1], nsf[q*4+2], nsf[q*4+3] };
            __builtin_nontemporal_store(v, (v4f*)(op + (size_t)i * nb + q * 4));
        }
        // leaky relu + fc_i (2x2)
        const float a0 = (y[0] >= 0.0f) ? y[0] : NEG_SLOPE * y[0];
        const float a1 = (y[1] >= 0.0f) ? y[1] : NEG_SLOPE * y[1];
        h0 = fmaf(W[210 + i*4 + 0], a0, fmaf(W[210 + i*4 + 1], a1, W[230 + i*2 + 0]));
        h1 = fmaf(W[210 + i*4 + 2], a0, fmaf(W[210 + i*4 + 3], a1, W[230 + i*2 + 1]));
    }

    // ---- fc10 (i = 4) ----
    const float o0 = fmaf(W[210 + 16 + 0], h0, fmaf(W[210 + 16 + 1], h1, W[230 + 8 + 0]));
    const float o1 = fmaf(W[210 + 16 + 2], h0, fmaf(W[210 + 16 + 3], h1, W[230 + 8 + 1]));
    v2f ov = { o0, o1 };
    __builtin_nontemporal_store(ov, (v2f*)(out_h + (size_t)n * 2));
}

extern "C" void kernel_launch(void* const* d_in, const int* in_sizes, int n_in,
                              void* d_out, int out_size, void* d_ws, size_t ws_size,
                              hipStream_t stream) {
    const float* x    = (const float*)d_in[0];
    const float* st   = (const float*)d_in[1];
    const float* fc1w = (const float*)d_in[2];
    const float* fc1b = (const float*)d_in[3];
    const float* fcw  = (const float*)d_in[4];
    const float* fcb  = (const float*)d_in[5];
    const float* A    = (const float*)d_in[6];
    const float* Bm   = (const float*)d_in[7];
    const float* C    = (const float*)d_in[8];
    const float* D    = (const float*)d_in[9];

    const int N = in_sizes[0] / 4;              // B = 1048576
    if (N <= 0) return;
    float* out_h = (float*)d_out;               // [N,2] first in return order
    float* out_s = out_h + (size_t)N * 2;       // [4,N,2,4,2] concatenated after

    dim3 block(256);
    dim3 grid((unsigned)((N + 255) / 256));
    hipLaunchKernelGGL(ssm_chain_kernel, grid, block, 0, stream,
                       x, st, fc1w, fc1b, fcw, fcb, A, Bm, C, D,
                       out_h, out_s, N);
}